// GraphConvolutionSparse_62062277427482
// MI455X (gfx1250) — compile-verified
//
#include <hip/hip_runtime.h>

// ---------------------------------------------------------------------------
// GraphConvolutionSparse on MI455X (gfx1250)
//   out = relu( A_sp @ ( dropout(X_sp) @ W ) )
// Strategy:
//   1) zero Xd[100000x256], h[100000x64] (ws) and out (d_out)
//   2) scatter dropout-scaled X nnz into dense Xd  (2M single-f32 atomics)
//   3) h = Xd @ W  dense GEMM via V_WMMA_F32_16X16X4_F32 (W staged in LDS)
//   4) out += per-edge a * h[col,:]  (coalesced f32 atomics, 64 thr/edge)
//   5) relu in place
// ---------------------------------------------------------------------------

#define N_NODES  100000
#define NNZ_FEAT 2000000
#define N_EDGES  1600000
#define IN_DIM   256
#define OUT_DIM  64

typedef float v2f __attribute__((ext_vector_type(2)));
typedef float v8f __attribute__((ext_vector_type(8)));

// ---------------------------------------------------------------- zero fill
__global__ void zero_f4_kernel(float4* __restrict__ p, long n4) {
    long i      = (long)blockIdx.x * blockDim.x + threadIdx.x;
    long stride = (long)gridDim.x * blockDim.x;
    float4 z = make_float4(0.f, 0.f, 0.f, 0.f);
    for (; i < n4; i += stride) p[i] = z;
}

// ------------------------------------------------- phase 1a: densify X (+dropout)
__global__ void scatter_x_kernel(const int* __restrict__ rows,
                                 const int* __restrict__ cols,
                                 const float* __restrict__ vals,
                                 const float* __restrict__ u,
                                 float* __restrict__ Xd) {
    int i = blockIdx.x * blockDim.x + threadIdx.x;
    if (i >= NNZ_FEAT) return;
    float ui = u[i];
    // keep iff floor(0.9 + u) >= 1, kept values scaled by 1/0.9
    if (floorf(0.9f + ui) >= 1.0f) {
        float v = vals[i] * (1.0f / 0.9f);
        atomicAdd(&Xd[(long)rows[i] * IN_DIM + cols[i]], v);
    }
}

// ------------------------------------------------- phase 1b: h = Xd @ W (WMMA f32)
// One wave computes a 16x64 strip of h. 8 waves / block -> 128 rows per block.
// A-frag (16x4 f32, 2 VGPR): lane<16 holds K=0,1 ; lane>=16 holds K=2,3 (row = lane%16)
// B-frag (4x16 f32, 2 VGPR): VGPR0 = K=0 (lanes 0-15) / K=2 (lanes 16-31); VGPR1 = K=1 / K=3
// C/D   (16x16 f32, 8 VGPR): VGPR r = row r (lanes 0-15) / row r+8 (lanes 16-31), N = lane%16
__global__ void __launch_bounds__(256)
gemm_xw_kernel(const float* __restrict__ Xd,
               const float* __restrict__ W,
               float* __restrict__ h) {
    __shared__ float Wl[IN_DIM * OUT_DIM];   // 64 KB, whole W

    // cooperative W load (coalesced float4)
    {
        const float4* Wv  = (const float4*)W;
        float4*       Wlv = (float4*)Wl;
        for (int i = threadIdx.x; i < (IN_DIM * OUT_DIM) / 4; i += 256)
            Wlv[i] = Wv[i];
    }
    __syncthreads();

    const int wave = threadIdx.x >> 5;
    const int lane = threadIdx.x & 31;
    const int hl   = lane >> 4;      // lane half: selects K pair
    const int lrow = lane & 15;

    const long strip = (long)blockIdx.x * 8 + wave;   // uniform per wave
    if (strip * 16 >= N_NODES) return;                 // whole-wave guard (EXEC stays all-1)
    const long m0 = strip * 16;

    v8f acc0 = {}, acc1 = {}, acc2 = {}, acc3 = {};

    // A row pointer for this lane: Xd[m0+lrow][k + 2*hl .. +1], 8B-aligned
    const v2f* __restrict__ ap =
        (const v2f*)(Xd + (m0 + lrow) * IN_DIM + 2 * hl);

    #pragma unroll 4
    for (int k = 0; k < IN_DIM; k += 4) {
        v2f a = ap[k >> 1];

        const int kb0 = (k + 2 * hl) * OUT_DIM + lrow;     // row K=2*hl
        const int kb1 = kb0 + OUT_DIM;                     // row K=2*hl+1

        v2f b0, b1, b2, b3;
        b0.x = Wl[kb0 +  0]; b0.y = Wl[kb1 +  0];
        b1.x = Wl[kb0 + 16]; b1.y = Wl[kb1 + 16];
        b2.x = Wl[kb0 + 32]; b2.y = Wl[kb1 + 32];
        b3.x = Wl[kb0 + 48]; b3.y = Wl[kb1 + 48];

        acc0 = __builtin_amdgcn_wmma_f32_16x16x4_f32(false, a, false, b0, (short)0, acc0, false, false);
        acc1 = __builtin_amdgcn_wmma_f32_16x16x4_f32(false, a, false, b1, (short)0, acc1, false, false);
        acc2 = __builtin_amdgcn_wmma_f32_16x16x4_f32(false, a, false, b2, (short)0, acc2, false, false);
        acc3 = __builtin_amdgcn_wmma_f32_16x16x4_f32(false, a, false, b3, (short)0, acc3, false, false);
    }

    // write back 16x64 strip
    float* __restrict__ hb = h + m0 * OUT_DIM;
    #pragma unroll
    for (int r = 0; r < 8; ++r) {
        const long m = r + 8 * hl;
        hb[m * OUT_DIM + lrow +  0] = acc0[r];
        hb[m * OUT_DIM + lrow + 16] = acc1[r];
        hb[m * OUT_DIM + lrow + 32] = acc2[r];
        hb[m * OUT_DIM + lrow + 48] = acc3[r];
    }
}

// ------------------------------------------------- phase 2: out += A @ h (scatter)
// 64 consecutive threads per edge: coalesced gather of h[col,:], coalesced atomics.
__global__ void spmm_adj_kernel(const int* __restrict__ rows,
                                const int* __restrict__ cols,
                                const float* __restrict__ vals,
                                const float* __restrict__ h,
                                float* __restrict__ out) {
    long t = (long)blockIdx.x * blockDim.x + threadIdx.x;
    long e = t >> 6;
    int  c = (int)(t & 63);
    if (e >= N_EDGES) return;
    int   r  = rows[e];
    int   cc = cols[e];
    float a  = vals[e];
    atomicAdd(&out[(long)r * OUT_DIM + c], a * h[(long)cc * OUT_DIM + c]);
}

// ------------------------------------------------- phase 3: relu in place
__global__ void relu_kernel(float* __restrict__ p, int n) {
    int i      = blockIdx.x * blockDim.x + threadIdx.x;
    int stride = gridDim.x * blockDim.x;
    for (; i < n; i += stride) p[i] = fmaxf(p[i], 0.0f);
}

// ---------------------------------------------------------------------------
extern "C" void kernel_launch(void* const* d_in, const int* in_sizes, int n_in,
                              void* d_out, int out_size, void* d_ws, size_t ws_size,
                              hipStream_t stream) {
    const int*   x_rows = (const int*)  d_in[0];
    const int*   x_cols = (const int*)  d_in[1];
    const float* x_vals = (const float*)d_in[2];
    const int*   a_rows = (const int*)  d_in[3];
    const int*   a_cols = (const int*)  d_in[4];
    const float* a_vals = (const float*)d_in[5];
    const float* u      = (const float*)d_in[6];
    const float* W      = (const float*)d_in[7];
    float*       out    = (float*)d_out;

    float* Xd = (float*)d_ws;                         // 100000*256 f32 = 102.4 MB
    float* h  = Xd + (long)N_NODES * IN_DIM;          // 100000*64  f32 =  25.6 MB

    // 1) zero Xd + h (contiguous in ws) and out
    const long n4_ws  = ((long)N_NODES * IN_DIM + (long)N_NODES * OUT_DIM) / 4;
    const long n4_out = ((long)N_NODES * OUT_DIM) / 4;
    zero_f4_kernel<<<4096, 256, 0, stream>>>((float4*)d_ws, n4_ws);
    zero_f4_kernel<<<2048, 256, 0, stream>>>((float4*)out, n4_out);

    // 2) densify dropout(X) into Xd
    scatter_x_kernel<<<(NNZ_FEAT + 255) / 256, 256, 0, stream>>>(
        x_rows, x_cols, x_vals, u, Xd);

    // 3) h = Xd @ W (WMMA). 6250 strips of 16 rows, 8 strips per block.
    const int n_strips = N_NODES / 16;                // 6250
    gemm_xw_kernel<<<(n_strips + 7) / 8, 256, 0, stream>>>(Xd, W, h);

    // 4) out += A @ h
    const long t_total = (long)N_EDGES * 64;
    spmm_adj_kernel<<<(int)((t_total + 255) / 256), 256, 0, stream>>>(
        a_rows, a_cols, a_vals, h, out);

    // 5) relu
    relu_kernel<<<2048, 256, 0, stream>>>(out, N_NODES * OUT_DIM);
}